// GRU_37864431681613
// MI455X (gfx1250) — compile-verified
//
#include <hip/hip_runtime.h>
#include <hip/hip_bf16.h>

// GRU (hidden=1), B=16384, T=4096.
// One lane per batch row; x staged through LDS with CDNA5 async
// global->LDS DMA loads (double buffered), fenced with s_wait_asynccnt.
// exp2 scale factors folded into the (uniform scalar) gate weights.

#define BDIM   256
#define T_LEN  4096
#define CHUNK  64                    // timesteps per LDS tile
#define STRIDE (CHUNK + 4)           // 68 floats = 272 B row stride: 16B aligned, spreads banks
#define LPT    (CHUNK / 4)           // 16 float4 async loads per thread per tile

static_assert(T_LEN % CHUNK == 0, "tile must divide T");

typedef __attribute__((address_space(3))) char as3_char;

// ---- CDNA5 async global->LDS copy, saddr form (16 bytes / lane) -------------
// gfx1250 MC: global_load_async_to_lds_b128 vdst(lds), voffset, s[base:base+1]
__device__ __forceinline__ void async_copy16(unsigned lds_off, unsigned g_off,
                                             const float* sbase) {
  asm volatile("global_load_async_to_lds_b128 %0, %1, %2"
               :: "v"(lds_off), "v"(g_off), "s"(sbase)
               : "memory");
}

#if __has_builtin(__builtin_amdgcn_s_wait_asynccnt)
#define WAIT_ASYNC(n) __builtin_amdgcn_s_wait_asynccnt(n)
#else
#define WAIT_ASYNC(n) asm volatile("s_wait_asynccnt %0" :: "n"(n) : "memory")
#endif

// ---- kernel -----------------------------------------------------------------
__global__ __launch_bounds__(BDIM)
void GRU_h1_stream_kernel(const float* __restrict__ x,
                          const float* __restrict__ w_ih,
                          const float* __restrict__ w_hh,
                          const float* __restrict__ b_ih,
                          const float* __restrict__ b_hh,
                          const float* __restrict__ fc_w,
                          const float* __restrict__ fc_b,
                          float* __restrict__ out)
{
  __shared__ __align__(16) float tile[2][BDIM * STRIDE];

  const int tid = threadIdx.x;
  const long long row0 = (long long)blockIdx.x * BDIM;
  const float* gblock = x + row0 * (long long)T_LEN;   // uniform -> SGPR pair

  // Uniform scalar weights with exp2 scaling folded in:
  //   sigmoid(a) = rcp(1 + exp2(KS*a)),  KS = -log2(e)
  //   tanh(a)    = 2*rcp(1 + exp2(KT*a)) - 1,  KT = -2*log2(e)
  const float KS = -1.442695040888963f;
  const float KT = -2.885390081777927f;
  const float wiR = KS * w_ih[0], whR = KS * w_hh[0], bR = KS * (b_ih[0] + b_hh[0]);
  const float wiZ = KS * w_ih[1], whZ = KS * w_hh[1], bZ = KS * (b_ih[1] + b_hh[1]);
  const float wiN = KT * w_ih[2], whN = KT * w_hh[2];
  const float bNI = KT * b_ih[2], bNH = KT * b_hh[2];

  // Per-thread constant pieces of the tile-slot mapping:
  //   slot idx = i*BDIM + tid ; r = i*16 + (tid>>4) ; cg = tid&15
  const unsigned rbase = (unsigned)(tid >> 4);
  const unsigned cg    = (unsigned)(tid & 15);

  // Issue one tile (256 rows x CHUNK cols) of coalesced async DMA loads.
  auto issue_tile = [&](int t0, int bsel) {
    #pragma unroll
    for (int i = 0; i < LPT; ++i) {
      unsigned r = (unsigned)i * 16u + rbase;
      unsigned g_off = (r * (unsigned)T_LEN + (unsigned)t0 + cg * 4u) << 2;  // bytes
      float* l = &tile[bsel][r * STRIDE + cg * 4u];
      unsigned lds_off = (unsigned)(size_t)(as3_char*)l;                     // LDS bytes
      async_copy16(lds_off, g_off, gblock);
    }
  };

  auto step = [&](float hh, float xv) -> float {
    float er = __builtin_amdgcn_exp2f(fmaf(whR, hh, fmaf(wiR, xv, bR)));
    float ez = __builtin_amdgcn_exp2f(fmaf(whZ, hh, fmaf(wiZ, xv, bZ)));
    float r  = __builtin_amdgcn_rcpf(1.0f + er);
    float z  = __builtin_amdgcn_rcpf(1.0f + ez);
    float an = fmaf(r, fmaf(whN, hh, bNH), fmaf(wiN, xv, bNI));   // KT * tanh-arg
    float en = __builtin_amdgcn_exp2f(an);
    float n  = fmaf(2.0f, __builtin_amdgcn_rcpf(1.0f + en), -1.0f);
    return fmaf(z, hh - n, n);                                    // (1-z)*n + z*h
  };

  issue_tile(0, 0);

  float h = 0.0f;
  int cur = 0;
  for (int t0 = 0; t0 < T_LEN; t0 += CHUNK) {
    if (t0 + CHUNK < T_LEN) {
      issue_tile(t0 + CHUNK, cur ^ 1);   // prefetch next tile into other buffer
      WAIT_ASYNC(LPT);                   // current tile's 16 loads complete
    } else {
      WAIT_ASYNC(0);
    }
    __syncthreads();                     // all waves' DMA for 'cur' visible

    const float* buf = &tile[cur][tid * STRIDE];
    #pragma unroll 4
    for (int c4 = 0; c4 < CHUNK / 4; ++c4) {
      float4 xv = *reinterpret_cast<const float4*>(buf + c4 * 4);
      h = step(h, xv.x);
      h = step(h, xv.y);
      h = step(h, xv.z);
      h = step(h, xv.w);
    }
    __syncthreads();                     // done reading 'cur' before it is re-filled
    cur ^= 1;
  }

  out[row0 + tid] = fmaf(fc_w[0], h, fc_b[0]);
}

// ---- host launch ------------------------------------------------------------
extern "C" void kernel_launch(void* const* d_in, const int* in_sizes, int n_in,
                              void* d_out, int out_size, void* d_ws, size_t ws_size,
                              hipStream_t stream) {
  const float* x    = (const float*)d_in[0];
  const float* w_ih = (const float*)d_in[1];
  const float* w_hh = (const float*)d_in[2];
  const float* b_ih = (const float*)d_in[3];
  const float* b_hh = (const float*)d_in[4];
  const float* fc_w = (const float*)d_in[5];
  const float* fc_b = (const float*)d_in[6];
  float* out = (float*)d_out;

  const int B = in_sizes[0] / T_LEN;          // 16384
  dim3 grid(B / BDIM), block(BDIM);
  hipLaunchKernelGGL(GRU_h1_stream_kernel, grid, block, 0, stream,
                     x, w_ih, w_hh, b_ih, b_hh, fc_w, fc_b, out);
}